// ChampferLoss_83210696393119
// MI455X (gfx1250) — compile-verified
//
#include <hip/hip_runtime.h>

typedef __attribute__((ext_vector_type(2))) float v2f;
typedef __attribute__((ext_vector_type(8))) float v8f;

#define BIGSQ   (1.0e8f * 1.0e8f)   // fl(1e16): reference's BIG minned-then-squared
#define SENTINEL 3.0e37f            // folded into |b|^2 for masked columns

// One wave owns 32 rows (two 16x4 A tiles) of point set A and computes, for
// each row, the min masked *squared* distance to every point of set B.
// Augmented-vector trick: A row = [-2ax,-2ay,-2az,|a|^2], B col = [bx,by,bz,1],
// so one V_WMMA_F32_16X16X4_F32 yields  |a|^2 - 2 a.b  for a 16x16 tile and
// the epilogue is just  d2 = c + |b|^2 ; clamp ; min.  sqrt is never needed
// (monotonic under min; reference squares the min afterwards anyway).
__global__ void chamfer_rowmin_kernel(const float* __restrict__ Apts,   // [B,Nrows,3]
                                      const int*   __restrict__ maskA,  // [B,Nrows] (used in reduce)
                                      const float* __restrict__ Bpts,   // [B,Ncols,3]
                                      const int*   __restrict__ maskB,  // [B,Ncols]
                                      float*       __restrict__ rowmin, // [B,Nrows] (min squared distance)
                                      int Nrows, int Ncols, int blocksPerBatch)
{
    const int wave = threadIdx.x >> 5;   // wave32
    const int lane = threadIdx.x & 31;
    const int half = lane >> 4;          // A/B: K={0,1} vs K={2,3}; D: rows r vs r+8
    const int l16  = lane & 15;

    const int b    = blockIdx.x / blocksPerBatch;
    const int nblk = blockIdx.x % blocksPerBatch;
    const int n0   = nblk * 128 + wave * 32;  // this wave's 32-row strip
    if (n0 >= Nrows) return;                  // wave-uniform: EXEC stays all-1s

    const float* Abase = Apts + (size_t)b * Nrows * 3;
    const float* Bbase = Bpts + (size_t)b * Ncols * 3;
    const int*   mBb   = maskB + (size_t)b * Ncols;

    // ---- Two augmented A tiles (rows n0+0..15, n0+16..31), 16x4 MxK layout:
    // lanes 0-15 hold K={0,1} = (-2ax,-2ay); lanes 16-31 hold K={2,3} = (-2az,|a|^2).
    v2f a[2];
    #pragma unroll
    for (int t = 0; t < 2; ++t) {
        const int arow = n0 + t * 16 + l16;
        const float* ap = Abase + (size_t)arow * 3;
        const float ax = ap[0], ay = ap[1], az = ap[2];
        const float a2 = ax * ax + ay * ay + az * az;
        a[t].x = half ? (-2.0f * az) : (-2.0f * ax);   // K=0 / K=2
        a[t].y = half ? a2 : (-2.0f * ay);             // K=1 / K=3 (|a|^2 slot)
    }

    float rmin[2][8];
    #pragma unroll
    for (int t = 0; t < 2; ++t)
        #pragma unroll
        for (int r = 0; r < 8; ++r) rmin[t][r] = BIGSQ;

    const int ntiles = Ncols >> 4;
    #pragma unroll 2
    for (int mt = 0; mt < ntiles; ++mt) {
        const int col = (mt << 4) + l16;         // this lane's D column
        const float* bp = Bbase + (size_t)col * 3;
        const float bx = bp[0], by = bp[1], bz = bp[2];
        const int   mb = mBb[col];
        // Column mask folded into |b|^2: masked columns can never win the min;
        // all-masked rows are clamped to BIGSQ at the end.
        const float b2 = (mb != 0) ? (bx * bx + by * by + bz * bz) : SENTINEL;

        // Augmented B tile, 4x16 KxN: lanes 0-15 K={0,1}=(bx,by);
        // lanes 16-31 K={2,3}=(bz, 1.0).
        v2f bm;
        bm.x = half ? bz : bx;
        bm.y = half ? 1.0f : by;

        #pragma unroll
        for (int t = 0; t < 2; ++t) {
            v8f c = {};
            c = __builtin_amdgcn_wmma_f32_16x16x4_f32(false, a[t], false, bm,
                                                      (short)0, c, false, false);
            #pragma unroll
            for (int r = 0; r < 8; ++r) {
                float d2 = c[r] + b2;            // (|a|^2 - 2a.b) + |b|^2
                d2 = fmaxf(d2, 0.0f);
                rmin[t][r] = fminf(rmin[t][r], d2);
            }
        }
    }

    // Min across the 16 lanes of each half (XOR masks < 16 stay in-half),
    // clamp to BIGSQ (reference: masked fill BIG, minned, then squared).
    #pragma unroll
    for (int t = 0; t < 2; ++t) {
        #pragma unroll
        for (int r = 0; r < 8; ++r) {
            float v = rmin[t][r];
            v = fminf(v, __shfl_xor(v, 1, 32));
            v = fminf(v, __shfl_xor(v, 2, 32));
            v = fminf(v, __shfl_xor(v, 4, 32));
            v = fminf(v, __shfl_xor(v, 8, 32));
            rmin[t][r] = fminf(v, BIGSQ);
        }
    }
    if (l16 == 0) {
        #pragma unroll
        for (int t = 0; t < 2; ++t)
            #pragma unroll
            for (int r = 0; r < 8; ++r)
                rowmin[(size_t)b * Nrows + n0 + t * 16 + r + 8 * half] = rmin[t][r];
    }
}

// out[b] = 0.5 * ( sum_n maskA*minsqA + sum_m maskB*minsqB )
__global__ void chamfer_reduce_kernel(const int*   __restrict__ maskA,
                                      const float* __restrict__ rminA, int N,
                                      const int*   __restrict__ maskB,
                                      const float* __restrict__ rminB, int M,
                                      float* __restrict__ out)
{
    __shared__ float sdata[256];
    const int b = blockIdx.x;
    float s = 0.0f;
    for (int i = threadIdx.x; i < N; i += blockDim.x) {
        if (maskA[(size_t)b * N + i] != 0)
            s += rminA[(size_t)b * N + i];
    }
    for (int i = threadIdx.x; i < M; i += blockDim.x) {
        if (maskB[(size_t)b * M + i] != 0)
            s += rminB[(size_t)b * M + i];
    }
    sdata[threadIdx.x] = s;
    __syncthreads();
    for (int off = 128; off > 0; off >>= 1) {
        if ((int)threadIdx.x < off) sdata[threadIdx.x] += sdata[threadIdx.x + off];
        __syncthreads();
    }
    if (threadIdx.x == 0) out[b] = 0.5f * sdata[0];
}

extern "C" void kernel_launch(void* const* d_in, const int* in_sizes, int n_in,
                              void* d_out, int out_size, void* d_ws, size_t ws_size,
                              hipStream_t stream) {
    const int*   o_w  = (const int*)  d_in[0];   // [B,N] int32
    const float* outs = (const float*)d_in[1];   // [B,N,3] f32
    const int*   t_w  = (const int*)  d_in[2];   // [B,M] int32
    const float* tgts = (const float*)d_in[3];   // [B,M,3] f32
    float* out = (float*)d_out;                  // [B]

    const int B = out_size;
    const int N = in_sizes[0] / B;
    const int M = in_sizes[2] / B;

    float* rminA = (float*)d_ws;                 // [B,N]
    float* rminB = rminA + (size_t)B * N;        // [B,M]

    const int bpbN = (N + 127) / 128;
    const int bpbM = (M + 127) / 128;
    dim3 blk(128);   // 4 waves of 32

    // min over targets for each output point
    chamfer_rowmin_kernel<<<B * bpbN, blk, 0, stream>>>(outs, o_w, tgts, t_w,
                                                        rminA, N, M, bpbN);
    // min over outputs for each target point (roles swapped)
    chamfer_rowmin_kernel<<<B * bpbM, blk, 0, stream>>>(tgts, t_w, outs, o_w,
                                                        rminB, M, N, bpbM);

    chamfer_reduce_kernel<<<B, 256, 0, stream>>>(o_w, rminA, N, t_w, rminB, M, out);
}